// message_passing_gnn_edges_gine_57363583205560
// MI455X (gfx1250) — compile-verified
//
#include <hip/hip_runtime.h>
#include <hip/hip_bf16.h>

typedef float v2f __attribute__((ext_vector_type(2)));
typedef float v8f __attribute__((ext_vector_type(8)));

#define NN 50000
#define EE 800000
#define GG 64
#define KC 64   // K chunk staged in LDS

// ---------------------------------------------------------------------------
// Fused GEMM:  C[M,NC] = act( concatK(A0,A1,A2) @ W[K,NC] + bias ) + residual
// All shapes compile-time -> fully unrolled K loop, immediate-offset loads.
// Block: 256 threads = 8 waves. Tile: 16 rows x 128 cols (wave = one 16x16
// V_WMMA_F32_16X16X4_F32 tile). Requires M%16==0, NC%128==0, K%KC==0,
// KSEG%KC==0 (each 64-wide chunk lies inside one concat segment).
// ---------------------------------------------------------------------------
template<int K, int KSEG, int NC>
__global__ __launch_bounds__(256)
void gemm_wmma_kernel(const float* __restrict__ A0, const float* __restrict__ A1,
                      const float* __restrict__ A2,
                      const float* __restrict__ W, const float* __restrict__ bias,
                      const float* __restrict__ residual, float* __restrict__ C,
                      int relu_flag)
{
    __shared__ float As[16][KC + 4];           // +4 pad -> conflict-free ds_load_b64

    const int tid  = threadIdx.x;
    const int wave = tid >> 5;
    const int lane = tid & 31;
    const int lc   = lane & 15;                // col-in-tile (B,C) / row-in-tile (A)
    const int half = lane >> 4;                // K-pair select (A,B) / M-half (C)
    const int kofs = half * 2;

    const int m0  = blockIdx.y * 16;
    const int n0  = blockIdx.x * 128 + wave * 16;
    const int col = n0 + lc;

    // per-lane W base: row (kofs), column col; all further offsets are
    // compile-time constants folded into global_load immediates
    const float* wbase = W + kofs * NC + col;

    v8f acc = {};

    #pragma unroll
    for (int kk = 0; kk < K; kk += KC) {
        constexpr_helper:;
        const int seg  = kk / KSEG;            // compile-time per unrolled chunk
        const int krem = kk - seg * KSEG;      // compile-time
        const float* Aseg = (seg == 0) ? A0 : ((seg == 1) ? A1 : A2);

        // cooperative load of A tile [16 x KC] into LDS (coalesced along K)
        #pragma unroll
        for (int i = 0; i < (16 * KC) / 256; ++i) {
            int idx = i * 256 + tid;
            int r   = idx >> 6;                // /KC
            int c   = idx & (KC - 1);
            As[r][c] = Aseg[(m0 + r) * KSEG + krem + c];
        }
        __syncthreads();

        #pragma unroll
        for (int k4 = 0; k4 < KC; k4 += 4) {
            // A 16x4 f32 frag: lanes 0-15 rows M, K = k4+{0,1}; lanes 16-31 K = k4+{2,3}
            v2f a;
            a.x = As[lc][k4 + kofs];
            a.y = As[lc][k4 + kofs + 1];
            // B 4x16 f32 frag: lanes 0-15 cols, rows K = k4+{0,1}; lanes 16-31 K = k4+{2,3}
            const float* wp = wbase + (kk + k4) * NC;   // constant offset
            v2f b;
            b.x = wp[0];
            b.y = wp[NC];
            acc = __builtin_amdgcn_wmma_f32_16x16x4_f32(
                false, a, false, b, (short)0, acc, false, false);
        }
        __syncthreads();
    }

    // epilogue: C/D layout — lane 0-15: M = m0+v ; lane 16-31: M = m0+8+v
    float bv = bias ? bias[col] : 0.0f;
    const int obase = (m0 + half * 8) * NC + col;
    #pragma unroll
    for (int v = 0; v < 8; ++v) {
        float val = acc[v] + bv;
        if (relu_flag) val = fmaxf(val, 0.0f);
        int o = obase + v * NC;
        if (residual) val += residual[o];
        C[o] = val;
    }
}

// ---------------------------------------------------------------------------
// Fused bidirectional message scatter. One wave per edge (grid-stride), lane
// covers 4 features (32*4 = 128). Edge encoder computed on the fly.
//   aggi[dst] += relu(x[src] + enc)   (parent_agg, src->dst)
//   aggo[src] += relu(x[dst] + enc)   (child_agg,  dst->src)
// aggi/aggo must be pre-initialized with x (gives the "x + agg" term).
// ---------------------------------------------------------------------------
__global__ __launch_bounds__(256)
void scatter_kernel(const float* __restrict__ x,
                    const int* __restrict__ src, const int* __restrict__ dst,
                    const float* __restrict__ eattr,
                    const float* __restrict__ We, const float* __restrict__ be,
                    float* __restrict__ aggi, float* __restrict__ aggo)
{
    const int lane = threadIdx.x & 31;
    const int wave = threadIdx.x >> 5;
    const int d0   = lane * 4;

    const float4 we  = *(const float4*)(We + d0);
    const float4 be4 = *(const float4*)(be + d0);

    const int nwaves = gridDim.x * 8;
    for (int e = blockIdx.x * 8 + wave; e < EE; e += nwaves) {
        int s = src[e];
        int d = dst[e];
        float ea = eattr[e];
        float4 enc;
        enc.x = fmaf(ea, we.x, be4.x);
        enc.y = fmaf(ea, we.y, be4.y);
        enc.z = fmaf(ea, we.z, be4.z);
        enc.w = fmaf(ea, we.w, be4.w);

        float4 xs = *(const float4*)(x + s * 128 + d0);
        float4 xd = *(const float4*)(x + d * 128 + d0);

        float* pi = aggi + d * 128 + d0;
        atomicAdd(pi + 0, fmaxf(xs.x + enc.x, 0.0f));
        atomicAdd(pi + 1, fmaxf(xs.y + enc.y, 0.0f));
        atomicAdd(pi + 2, fmaxf(xs.z + enc.z, 0.0f));
        atomicAdd(pi + 3, fmaxf(xs.w + enc.w, 0.0f));

        float* po = aggo + s * 128 + d0;
        atomicAdd(po + 0, fmaxf(xd.x + enc.x, 0.0f));
        atomicAdd(po + 1, fmaxf(xd.y + enc.y, 0.0f));
        atomicAdd(po + 2, fmaxf(xd.z + enc.z, 0.0f));
        atomicAdd(po + 3, fmaxf(xd.w + enc.w, 0.0f));
    }
}

// ---------------------------------------------------------------------------
// Pooling
// ---------------------------------------------------------------------------
__device__ __forceinline__ void atomicMaxF(float* addr, float val)
{
    unsigned int* ia = (unsigned int*)addr;
    unsigned int old = *ia;
    while (__uint_as_float(old) < val) {
        unsigned int assumed = old;
        old = atomicCAS(ia, assumed, __float_as_uint(val));
        if (old == assumed) break;
    }
}

__global__ void pool_init_kernel(float* gmax, float* gsum, float* gcnt)
{
    int i = blockIdx.x * blockDim.x + threadIdx.x;
    if (i < GG * 256) { gmax[i] = -3.402823466e38f; gsum[i] = 0.0f; }
    if (i < GG) gcnt[i] = 0.0f;
}

__global__ void pool_count_kernel(const int* __restrict__ batch, float* gcnt)
{
    int i = blockIdx.x * blockDim.x + threadIdx.x;
    if (i < NN) atomicAdd(&gcnt[batch[i]], 1.0f);
}

// thread = feature (0..255), block = 256-node chunk; batch is sorted so each
// thread does run-length local max/sum and flushes one atomic pair per run.
__global__ __launch_bounds__(256)
void pool_kernel(const float* __restrict__ xc, const int* __restrict__ batch,
                 float* __restrict__ gmax, float* __restrict__ gsum)
{
    const int f  = threadIdx.x;
    const int nb = blockIdx.x * 256;
    const int ne = min(nb + 256, NN);
    if (nb >= NN) return;

    int   curg = batch[nb];
    float m = -3.402823466e38f, s = 0.0f;
    for (int n = nb; n < ne; ++n) {
        int g = batch[n];                        // uniform across the block
        if (g != curg) {
            atomicMaxF(&gmax[curg * 256 + f], m);
            atomicAdd(&gsum[curg * 256 + f], s);
            curg = g; m = -3.402823466e38f; s = 0.0f;
        }
        float v = xc[n * 256 + f];
        m = fmaxf(m, v);
        s += v;
    }
    atomicMaxF(&gmax[curg * 256 + f], m);
    atomicAdd(&gsum[curg * 256 + f], s);
}

__global__ void pool_final_kernel(const float* __restrict__ gmax,
                                  const float* __restrict__ gsum,
                                  const float* __restrict__ gcnt,
                                  float* __restrict__ out)
{
    int i = blockIdx.x * blockDim.x + threadIdx.x;
    if (i >= GG * 512) return;
    int g = i >> 9;
    int f = i & 511;
    if (f < 256) out[i] = gmax[g * 256 + f];
    else         out[i] = gsum[g * 256 + (f - 256)] / fmaxf(gcnt[g], 1.0f);
}

// ---------------------------------------------------------------------------
extern "C" void kernel_launch(void* const* d_in, const int* in_sizes, int n_in,
                              void* d_out, int out_size, void* d_ws, size_t ws_size,
                              hipStream_t stream)
{
    const float* nodes  = (const float*)d_in[0];   // [N,256]
    const int*   edges  = (const int*)  d_in[1];   // [2,E] flat
    const float* eattr  = (const float*)d_in[2];   // [E]
    const int*   batch  = (const int*)  d_in[3];   // [N]
    const float* W_init = (const float*)d_in[4];
    const float* b_init = (const float*)d_in[5];
    const float* W_edge = (const float*)d_in[6];
    const float* b_edge = (const float*)d_in[7];
    const float* W1a = (const float*)d_in[8];  const float* b1a = (const float*)d_in[9];
    const float* W2a = (const float*)d_in[10]; const float* b2a = (const float*)d_in[11];
    const float* W1b = (const float*)d_in[12]; const float* b1b = (const float*)d_in[13];
    const float* W2b = (const float*)d_in[14]; const float* b2b = (const float*)d_in[15];
    const float* Wf  = (const float*)d_in[16]; const float* bf  = (const float*)d_in[17];
    const float* Wf2 = (const float*)d_in[18]; const float* bf2 = (const float*)d_in[19];
    const float* Wc  = (const float*)d_in[20]; const float* bc  = (const float*)d_in[21];

    const int* src = edges;        // edges[0]
    const int* dst = edges + EE;   // edges[1]

    // workspace carve-up (floats)
    float* ws   = (float*)d_ws;
    float* x    = ws;                              // N*128
    float* xn   = x    + (size_t)NN * 128;         // N*128
    float* aggi = xn   + (size_t)NN * 128;         // N*128
    float* aggo = aggi + (size_t)NN * 128;         // N*128
    float* tmp  = aggo + (size_t)NN * 128;         // N*256 (hidden / conv out)
    float* gmax = tmp  + (size_t)NN * 256;         // G*256
    float* gsum = gmax + (size_t)GG * 256;         // G*256
    float* gcnt = gsum + (size_t)GG * 256;         // G

    const dim3 blk(256);
    const dim3 gD(1, NN / 16);    // NC = 128 GEMMs
    const dim3 g2D(2, NN / 16);   // NC = 256 GEMMs

    // x = nodes @ W_init + b_init                       [N,256]x[256,128]
    gemm_wmma_kernel<256, 256, 128><<<gD, blk, 0, stream>>>(
        nodes, nullptr, nullptr, W_init, b_init, nullptr, x, 0);

    for (int it = 0; it < 2; ++it) {
        // agg = x  (the "x + agg" term of GINE with eps=0)
        hipMemcpyAsync(aggi, x, (size_t)NN * 128 * sizeof(float),
                       hipMemcpyDeviceToDevice, stream);
        hipMemcpyAsync(aggo, x, (size_t)NN * 128 * sizeof(float),
                       hipMemcpyDeviceToDevice, stream);
        // scatter both directions, edge-encoder fused
        scatter_kernel<<<2048, blk, 0, stream>>>(x, src, dst, eattr,
                                                 W_edge, b_edge, aggi, aggo);
        // fi = mlp_1(aggi): tmp = relu(aggi@W1a+b1a); aggi <- tmp@W2a+b2a
        gemm_wmma_kernel<128, 128, 128><<<gD, blk, 0, stream>>>(
            aggi, nullptr, nullptr, W1a, b1a, nullptr, tmp, 1);
        gemm_wmma_kernel<128, 128, 128><<<gD, blk, 0, stream>>>(
            tmp, nullptr, nullptr, W2a, b2a, nullptr, aggi, 0);
        // fo = mlp_2(aggo): tmp = relu(aggo@W1b+b1b); aggo <- tmp@W2b+b2b
        gemm_wmma_kernel<128, 128, 128><<<gD, blk, 0, stream>>>(
            aggo, nullptr, nullptr, W1b, b1b, nullptr, tmp, 1);
        gemm_wmma_kernel<128, 128, 128><<<gD, blk, 0, stream>>>(
            tmp, nullptr, nullptr, W2b, b2b, nullptr, aggo, 0);
        // hidden = relu([x, fi, fo] @ Wf + bf)          [N,384]x[384,256]
        gemm_wmma_kernel<384, 128, 256><<<g2D, blk, 0, stream>>>(
            x, aggi, aggo, Wf, bf, nullptr, tmp, 1);
        // xn = x + (hidden @ Wf2 + bf2)                 [N,256]x[256,128] + residual
        gemm_wmma_kernel<256, 256, 128><<<gD, blk, 0, stream>>>(
            tmp, nullptr, nullptr, Wf2, bf2, x, xn, 0);
        float* t = x; x = xn; xn = t;   // ITERS fixed -> deterministic
    }

    // conv1: tmp = x @ Wc + bc                          [N,128]x[128,256]
    gemm_wmma_kernel<128, 128, 256><<<g2D, blk, 0, stream>>>(
        x, nullptr, nullptr, Wc, bc, nullptr, tmp, 0);

    // global max / avg pooling per graph
    pool_init_kernel <<<(GG * 256 + 255) / 256, blk, 0, stream>>>(gmax, gsum, gcnt);
    pool_count_kernel<<<(NN + 255) / 256,       blk, 0, stream>>>(batch, gcnt);
    pool_kernel      <<<(NN + 255) / 256,       blk, 0, stream>>>(tmp, batch, gmax, gsum);
    pool_final_kernel<<<(GG * 512 + 255) / 256, blk, 0, stream>>>(gmax, gsum, gcnt,
                                                                  (float*)d_out);
}